// InductiveModel_52759378264194
// MI455X (gfx1250) — compile-verified
//
#include <hip/hip_runtime.h>
#include <cstdint>
#include <cstddef>

typedef __attribute__((ext_vector_type(2))) float v2f;
typedef __attribute__((ext_vector_type(8))) float v8f;

#define NEG_SLOPE 0.2f

__device__ __forceinline__ void atomic_max_f32(float* addr, float v) {
  // Positive floats order as ints; negative floats reverse-order as uints.
  if (v >= 0.0f) atomicMax((int*)addr, __float_as_int(v));
  else           atomicMin((unsigned int*)addr, __float_as_uint(v));
}

// ---------------------------------------------------------------------------
// C[M,Nc] = act(A[M,K]) * B[K,Nc], act = FUSE ? relu(a + biasA[col]) : a
// One wave computes a 16x64 output tile with V_WMMA_F32_16X16X4_F32.
// Requires: M%16==0, K%4==0, Nc%64==0 (holds: M=50000, K in {128,192}, Nc=192)
// A fragment (16x4 f32): lane m=lane&15 holds K = 2*(lane>>4) + {0,1}
// B fragment (4x16 f32): lane n=lane&15 holds the same K split
// C/D (16x16 f32, 8 VGPRs): row = r + 8*(lane>>4), col = lane&15
// ---------------------------------------------------------------------------
template <bool FUSE>
__global__ void gat_gemm_wmma(const float* __restrict__ A, const float* __restrict__ B,
                              const float* __restrict__ biasA, float* __restrict__ C,
                              int M, int K, int Nc) {
  const int lane  = threadIdx.x & 31;
  const int wave  = (int)((blockIdx.x * blockDim.x + threadIdx.x) >> 5);
  const int tilesN = Nc >> 6;           // 64-wide column tiles
  const int tilesM = M >> 4;
  if (wave >= tilesM * tilesN) return;  // wave-uniform: EXEC stays all-1s
  const int tm = wave / tilesN;
  const int tn = wave % tilesN;
  const int lo = lane & 15;
  const int hi = lane >> 4;

  // Running pointers: no per-iteration 64-bit multiplies in the k-loop.
  const float* aptr = A + (size_t)(tm * 16 + lo) * K + 2 * hi;   // 8B aligned
  const float* bptr = B + (size_t)(2 * hi) * Nc + tn * 64 + lo;  // row ka
  const float* bias = biasA + 2 * hi;
  const size_t bstep = (size_t)4 * Nc;

  v8f acc0 = {}; v8f acc1 = {}; v8f acc2 = {}; v8f acc3 = {};
  for (int k = 0; k < K; k += 4) {
    v2f a = *(const v2f*)aptr;
    if (FUSE) {
      const v2f bb = *(const v2f*)bias;
      a.x = fmaxf(a.x + bb.x, 0.0f);
      a.y = fmaxf(a.y + bb.y, 0.0f);
      bias += 4;
    }
    const float* b0p = bptr;         // row ka
    const float* b1p = bptr + Nc;    // row ka+1
    v2f b0; b0.x = b0p[0];  b0.y = b1p[0];
    v2f b1; b1.x = b0p[16]; b1.y = b1p[16];
    v2f b2; b2.x = b0p[32]; b2.y = b1p[32];
    v2f b3; b3.x = b0p[48]; b3.y = b1p[48];
    acc0 = __builtin_amdgcn_wmma_f32_16x16x4_f32(false, a, false, b0, (short)0, acc0, false, false);
    acc1 = __builtin_amdgcn_wmma_f32_16x16x4_f32(false, a, false, b1, (short)0, acc1, false, false);
    acc2 = __builtin_amdgcn_wmma_f32_16x16x4_f32(false, a, false, b2, (short)0, acc2, false, false);
    acc3 = __builtin_amdgcn_wmma_f32_16x16x4_f32(false, a, false, b3, (short)0, acc3, false, false);
    aptr += 4;
    bptr += bstep;
  }
  float* Crow = C + tn * 64 + lo;
  #pragma unroll
  for (int r = 0; r < 8; ++r) {
    const size_t row = (size_t)(tm * 16 + r + 8 * hi) * Nc;
    Crow[row + 0]  = acc0[r];
    Crow[row + 16] = acc1[r];
    Crow[row + 32] = acc2[r];
    Crow[row + 48] = acc3[r];
  }
}

// ---------------------------------------------------------------------------
// a_s[n,h] = <h[n,h,:], att_src[h,:]>, a_d likewise. One wave per (n,h).
// ---------------------------------------------------------------------------
__global__ void gat_attn(const float* __restrict__ h, const float* __restrict__ att_src,
                         const float* __restrict__ att_dst, float* __restrict__ a_s,
                         float* __restrict__ a_d, int N, int H, int C) {
  const int wave = (int)((blockIdx.x * blockDim.x + threadIdx.x) >> 5);
  const int lane = threadIdx.x & 31;
  if (wave >= N * H) return;
  const int n = wave / H, hh = wave % H;
  const float* hp = h + (size_t)n * H * C + (size_t)hh * C;
  float s0 = 0.0f, s1 = 0.0f;
  for (int c = lane; c < C; c += 32) {
    const float v = hp[c];
    s0 += v * att_src[hh * C + c];
    s1 += v * att_dst[hh * C + c];
  }
  #pragma unroll
  for (int off = 16; off > 0; off >>= 1) {
    s0 += __shfl_down(s0, off);
    s1 += __shfl_down(s1, off);
  }
  if (lane == 0) { a_s[wave] = s0; a_d[wave] = s1; }
}

// ---------------------------------------------------------------------------
__global__ void gat_init(float* __restrict__ out, float* __restrict__ emax,
                         float* __restrict__ denom, int N, int HC, int H) {
  const int i = blockIdx.x * blockDim.x + threadIdx.x;
  if (i < N * HC) out[i] = 0.0f;
  if (i < N * H) { emax[i] = -__builtin_inff(); denom[i] = 0.0f; }
}

__device__ __forceinline__ void edge_sd(const long long* __restrict__ ei, int e, int E,
                                        int& s, int& d) {
  if (e < E) { s = (int)ei[e]; d = (int)ei[(size_t)E + e]; }
  else       { s = d = e - E; }   // self-loops appended after the E real edges
}

// Pass A: segment max of leaky_relu(a_s[src]+a_d[dst]) over dst.
__global__ void gat_edge_max(const long long* __restrict__ ei, int E, int N,
                             const float* __restrict__ a_s, const float* __restrict__ a_d,
                             float* __restrict__ emax, int H) {
  const int e = blockIdx.x * blockDim.x + threadIdx.x;
  if (e >= E + N) return;
  int s, d; edge_sd(ei, e, E, s, d);
  for (int h = 0; h < H; ++h) {
    float v = a_s[s * H + h] + a_d[d * H + h];
    v = (v > 0.0f) ? v : NEG_SLOPE * v;
    atomic_max_f32(&emax[d * H + h], v);
  }
}

// Pass B: e = exp(leaky - emax[dst]); cache per-edge e; denom[dst] += e.
__global__ void gat_edge_exp(const long long* __restrict__ ei, int E, int N,
                             const float* __restrict__ a_s, const float* __restrict__ a_d,
                             const float* __restrict__ emax, float* __restrict__ ework,
                             float* __restrict__ denom, int H) {
  const int e = blockIdx.x * blockDim.x + threadIdx.x;
  if (e >= E + N) return;
  int s, d; edge_sd(ei, e, E, s, d);
  for (int h = 0; h < H; ++h) {
    float v = a_s[s * H + h] + a_d[d * H + h];
    v = (v > 0.0f) ? v : NEG_SLOPE * v;
    const float ex = __expf(v - emax[d * H + h]);
    ework[(size_t)e * H + h] = ex;
    atomicAdd(&denom[d * H + h], ex);
  }
}

// Pass C: out[dst,h,c] += h[src,h,c] * (e / denom[dst,h]).
// 64 threads per edge; thread k owns channel k of every head -> coalesced.
__global__ void gat_edge_agg(const long long* __restrict__ ei, int E, int N,
                             const float* __restrict__ hfeat, const float* __restrict__ ework,
                             const float* __restrict__ denom, float* __restrict__ out,
                             int H, int C) {
  const long long gid = (long long)blockIdx.x * blockDim.x + threadIdx.x;
  const int e = (int)(gid >> 6);
  const int c = (int)(gid & 63);
  if (e >= E + N) return;
  int s, d; edge_sd(ei, e, E, s, d);
  const size_t sb = (size_t)s * H * C, db = (size_t)d * H * C;
  for (int h = 0; h < H; ++h) {
    const float alpha = ework[(size_t)e * H + h] / denom[d * H + h];
    atomicAdd(&out[db + (size_t)h * C + c], hfeat[sb + (size_t)h * C + c] * alpha);
  }
}

// Final: mean over heads + bias -> d_out [N, C]
__global__ void gat_finalize(const float* __restrict__ out2, const float* __restrict__ b2,
                             float* __restrict__ y, int N, int H, int C) {
  const int i = blockIdx.x * blockDim.x + threadIdx.x;
  if (i >= N * C) return;
  const int n = i / C, c = i % C;
  float s = 0.0f;
  for (int h = 0; h < H; ++h) s += out2[(size_t)n * H * C + (size_t)h * C + c];
  y[i] = s * (1.0f / (float)H) + b2[c];
}

// ---------------------------------------------------------------------------
extern "C" void kernel_launch(void* const* d_in, const int* in_sizes, int n_in,
                              void* d_out, int out_size, void* d_ws, size_t ws_size,
                              hipStream_t stream) {
  const float*     x        = (const float*)d_in[0];
  const long long* ei       = (const long long*)d_in[1];
  const float*     W1       = (const float*)d_in[2];
  const float*     att_src1 = (const float*)d_in[3];
  const float*     att_dst1 = (const float*)d_in[4];
  const float*     b1       = (const float*)d_in[5];
  const float*     W2       = (const float*)d_in[6];
  const float*     att_src2 = (const float*)d_in[7];
  const float*     att_dst2 = (const float*)d_in[8];
  const float*     b2       = (const float*)d_in[9];
  float* y = (float*)d_out;

  const int F_in = 128, H = 3, C = 64, HC = H * C;   // 192
  const int N  = in_sizes[0] / F_in;                 // 50000
  const int E  = in_sizes[1] / 2;                    // 800000
  const int EN = E + N;                              // edges + self-loops

  float* ws = (float*)d_ws;
  const size_t NF = (size_t)N * HC;
  float* h1    = ws;                 // [N, 192]
  float* out1  = ws + NF;            // [N, 192]
  float* h2    = ws + 2 * NF;        // [N, 192]
  float* out2  = h1;                 // alias: h1 is dead after layer-1 agg
  float* ework = ws + 3 * NF;        // [(E+N), 3]
  float* a_s   = ework + (size_t)EN * H;
  float* a_d   = a_s + (size_t)N * H;
  float* emax  = a_d + (size_t)N * H;
  float* denom = emax + (size_t)N * H;

  const int TPB = 256;
  const int gemm_tasks = (N / 16) * (HC / 64);                  // waves
  const int gemm_grid  = (gemm_tasks * 32 + TPB - 1) / TPB;
  const int attn_grid  = (N * H * 32 + TPB - 1) / TPB;
  const int init_grid  = (N * HC + TPB - 1) / TPB;
  const int edge_grid  = (EN + TPB - 1) / TPB;
  const long long aggw = (long long)EN * 64;
  const int agg_grid   = (int)((aggw + TPB - 1) / TPB);
  const int fin_grid   = (N * C + TPB - 1) / TPB;

  // ---- Layer 1: h1 = x @ W1 ; attention ; softmax ; aggregate -> out1
  gat_gemm_wmma<false><<<gemm_grid, TPB, 0, stream>>>(x, W1, nullptr, h1, N, F_in, HC);
  gat_attn<<<attn_grid, TPB, 0, stream>>>(h1, att_src1, att_dst1, a_s, a_d, N, H, C);
  gat_init<<<init_grid, TPB, 0, stream>>>(out1, emax, denom, N, HC, H);
  gat_edge_max<<<edge_grid, TPB, 0, stream>>>(ei, E, N, a_s, a_d, emax, H);
  gat_edge_exp<<<edge_grid, TPB, 0, stream>>>(ei, E, N, a_s, a_d, emax, ework, denom, H);
  gat_edge_agg<<<agg_grid, TPB, 0, stream>>>(ei, E, N, h1, ework, denom, out1, H, C);

  // ---- Layer 2: h2 = relu(out1 + b1) @ W2 (bias+relu fused into A load)
  gat_gemm_wmma<true><<<gemm_grid, TPB, 0, stream>>>(out1, W2, b1, h2, N, HC, HC);
  gat_attn<<<attn_grid, TPB, 0, stream>>>(h2, att_src2, att_dst2, a_s, a_d, N, H, C);
  gat_init<<<init_grid, TPB, 0, stream>>>(out2, emax, denom, N, HC, H);
  gat_edge_max<<<edge_grid, TPB, 0, stream>>>(ei, E, N, a_s, a_d, emax, H);
  gat_edge_exp<<<edge_grid, TPB, 0, stream>>>(ei, E, N, a_s, a_d, emax, ework, denom, H);
  gat_edge_agg<<<agg_grid, TPB, 0, stream>>>(ei, E, N, h2, ework, denom, out2, H, C);

  // ---- mean over heads + bias
  gat_finalize<<<fin_grid, TPB, 0, stream>>>(out2, b2, y, N, H, C);
}